// SpatialTemporalAggregator_1692217115134
// MI455X (gfx1250) — compile-verified
//
#include <hip/hip_runtime.h>
#include <hip/hip_bf16.h>

typedef __attribute__((ext_vector_type(2))) float v2f;
typedef __attribute__((ext_vector_type(8))) float v8f;

#define HEADS 8
#define NB    4      // batch
#define TT    16     // total frames
#define TTEM  8
#define DCH   512
#define DH    64
#define HH    64
#define WW    64
#define KH    8
#define KW    8
#define KK    512    // ts*ph*pw (GEMM K)
#define MM    256    // h0*w0   (GEMM M)
#define PLANE 4096   // 64*64

// jax.image.resize 'linear' 16->64 (half-pixel centers, edge clamp).
__device__ __forceinline__ void lin_coeff(int o, int& i0, int& i1, float& t) {
    float f  = o * 0.25f - 0.375f;
    float fl = floorf(f);
    t = f - fl;
    int i = (int)fl;
    int j = i + 1;
    i0 = i < 0 ? 0 : (i > 15 ? 15 : i);
    i1 = j < 0 ? 0 : (j > 15 ? 15 : j);
}

// ---------------------------------------------------------------------------
// Kernel 1: 8x8 average pool of frames 8..15 -> pooled[(head*NB+b)][k=ts*64+cell][dh]
// 1,048,576 threads; float4 row loads (contiguous 256B per 8 lanes).
// ---------------------------------------------------------------------------
__global__ void pool_kernel(const float* __restrict__ x, float* __restrict__ pooled) {
    int tid  = blockIdx.x * blockDim.x + threadIdx.x;   // 2^20 threads
    int cell = tid & 63;               // py*8+px
    int c    = (tid >> 6) & 511;       // channel = head*64+dh
    int ts   = (tid >> 15) & 7;        // spatial frame 0..7
    int b    = tid >> 18;              // batch

    int px = cell & 7, py = cell >> 3;
    int bt = b * TT + TTEM + ts;       // frames 8..15
    const float* p = x + ((size_t)bt * DCH + c) * PLANE + (py * KH) * WW + px * KW;

    float s = 0.f;
#pragma unroll
    for (int ky = 0; ky < KH; ++ky) {
        const float4* r = reinterpret_cast<const float4*>(p + ky * WW);
        float4 a = r[0], q = r[1];
        s += a.x + a.y + a.z + a.w + q.x + q.y + q.z + q.w;
    }
    int head = c >> 6, dh = c & 63;
    int hb = head * NB + b;
    pooled[((size_t)hb * KK + (ts * 64 + cell)) * DH + dh] = s * (1.0f / 64.0f);
}

// ---------------------------------------------------------------------------
// Kernel 2: bilinear upsample temporal mask (head,b,t,16,16) -> (head,b,t,64,64)
// ---------------------------------------------------------------------------
__global__ void attn_up_kernel(const float* __restrict__ mask, float* __restrict__ up) {
    int tid = blockIdx.x * blockDim.x + threadIdx.x;    // 2^20 threads
    int xx = tid & 63, yy = (tid >> 6) & 63;
    int plane = tid >> 12;                              // ((head*4+b)*8+t)
    const float* mp = mask + (size_t)plane * 256;

    int iy0, iy1, ix0, ix1; float ty, tx;
    lin_coeff(yy, iy0, iy1, ty);
    lin_coeff(xx, ix0, ix1, tx);
    float v00 = mp[iy0 * 16 + ix0], v01 = mp[iy0 * 16 + ix1];
    float v10 = mp[iy1 * 16 + ix0], v11 = mp[iy1 * 16 + ix1];
    up[tid] = (1.f - ty) * ((1.f - tx) * v00 + tx * v01)
            +        ty  * ((1.f - tx) * v10 + tx * v11);
}

// ---------------------------------------------------------------------------
// Kernel 3: 32x GEMM (256x512 @ 512x64) with V_WMMA_F32_16X16X4_F32.
// One wave32 per 16x16 D tile; K-loop step 4.
// Output stored transposed: spa_t[(hb*64+dh)*256 + m]  (== reference transpose).
// A layout (ISA 16x4 f32): lanes 0-15 M=lane {K=0,K=1}; lanes 16-31 {K=2,K=3}.
// B layout (4x16 f32): VGPR0 row K=0|2 across lanes, VGPR1 row K=1|3.
// D layout: lane<16 -> VGPRj = D[M=j][N=lane]; lane>=16 -> D[M=j+8][N=lane-16].
// ---------------------------------------------------------------------------
__global__ void spa_wmma_kernel(const float* __restrict__ A,
                                const float* __restrict__ Bp,
                                float* __restrict__ spa_t) {
    int wave = (int)((blockIdx.x * blockDim.x + threadIdx.x) >> 5); // 0..2047
    int lane = threadIdx.x & 31;
    int hb   = wave >> 6;            // 0..31   (head*4+b)
    int rem  = wave & 63;
    int m0   = (rem >> 2) << 4;      // 16 M tiles
    int n0   = (rem & 3) << 4;       // 4  N tiles
    int half = lane >> 4;            // 0|1
    int lm   = lane & 15;

    const float* Ab = A  + (size_t)hb * MM * KK + (size_t)(m0 + lm) * KK + half * 2;
    const float* Bb = Bp + (size_t)hb * KK * DH + (size_t)(half * 2) * DH + n0 + lm;

    v8f acc = {};
#if __has_builtin(__builtin_amdgcn_wmma_f32_16x16x4_f32)
    for (int k = 0; k < KK; k += 4) {
        v2f a = *reinterpret_cast<const v2f*>(Ab + k);   // A[m][k+half*2 .. +1], 8B aligned
        v2f b;
        b.x = Bb[(size_t)k * DH];                        // B[k+half*2    ][n]
        b.y = Bb[(size_t)k * DH + DH];                   // B[k+half*2 + 1][n]
        acc = __builtin_amdgcn_wmma_f32_16x16x4_f32(
            /*neg_a=*/false, a, /*neg_b=*/false, b,
            /*c_mod=*/(short)0, acc, /*reuse_a=*/false, /*reuse_b=*/false);
    }
#else
#warning "wmma_f32_16x16x4_f32 builtin not available: scalar fallback in use"
    // Scalar fallback computing the identical lane-mapped result.
    for (int k = 0; k < KK; k += 4) {
        for (int kk = 0; kk < 4; ++kk) {
#pragma unroll
            for (int j = 0; j < 8; ++j) {
                int m = m0 + j + half * 8;
                int n = n0 + lm;
                acc[j] += A[(size_t)hb * MM * KK + (size_t)m * KK + k + kk]
                        * Bp[(size_t)hb * KK * DH + (size_t)(k + kk) * DH + n];
            }
        }
    }
#endif

    float* o = spa_t + ((size_t)hb * DH + n0 + lm) * MM + m0 + half * 8;
#pragma unroll
    for (int j = 0; j < 8; ++j) o[j] = acc[j];
}

// ---------------------------------------------------------------------------
// Kernel 4: final fuse. 1 thread per output element (b,c=head*64+dh,y,x).
// out = x[t0] + sum_t w_t * x[t] + bilerp(spa_t[hb][dh])
// ---------------------------------------------------------------------------
__global__ void final_kernel(const float* __restrict__ x,
                             const float* __restrict__ attn_up,
                             const float* __restrict__ spa_t,
                             float* __restrict__ out) {
    int tid = blockIdx.x * blockDim.x + threadIdx.x;   // 2^23 threads
    int xx = tid & 63, yy = (tid >> 6) & 63;
    int c  = (tid >> 12) & 511;
    int b  = tid >> 21;
    int head = c >> 6, dh = c & 63;
    int hb = head * NB + b;

    const float* xp = x + ((size_t)(b * TT) * DCH + c) * PLANE + yy * WW + xx;
    const float* au = attn_up + ((size_t)hb * TTEM) * PLANE + yy * WW + xx;

    float xv0 = xp[0];
    float acc = xv0 * (1.0f + au[0]);           // self (t=0) + weighted t=0
#pragma unroll
    for (int t = 1; t < TTEM; ++t)
        acc += au[(size_t)t * PLANE] * xp[(size_t)t * DCH * PLANE];

    int iy0, iy1, ix0, ix1; float ty, tx;
    lin_coeff(yy, iy0, iy1, ty);
    lin_coeff(xx, ix0, ix1, tx);
    const float* sp = spa_t + ((size_t)hb * DH + dh) * MM;
    float v00 = sp[iy0 * 16 + ix0], v01 = sp[iy0 * 16 + ix1];
    float v10 = sp[iy1 * 16 + ix0], v11 = sp[iy1 * 16 + ix1];
    float spav = (1.f - ty) * ((1.f - tx) * v00 + tx * v01)
               +        ty  * ((1.f - tx) * v10 + tx * v11);

    out[tid] = acc + spav;
}

extern "C" void kernel_launch(void* const* d_in, const int* in_sizes, int n_in,
                              void* d_out, int out_size, void* d_ws, size_t ws_size,
                              hipStream_t stream) {
    const float* x  = (const float*)d_in[0];   // (64, 512, 64, 64)
    const float* mt = (const float*)d_in[1];   // (8, 4, 8, 16, 16)
    const float* ms = (const float*)d_in[2];   // (8, 4, 256, 512)
    float* out = (float*)d_out;                // (4, 512, 64, 64)

    // Workspace layout (10 MB total):
    float* pooled  = (float*)d_ws;             // 1,048,576 f32 : (32, 512, 64)
    float* spa_t   = pooled + (1 << 20);       //   524,288 f32 : (32, 64, 256)
    float* attn_up = spa_t + (1 << 19);        // 1,048,576 f32 : (8,4,8,64,64)

    hipLaunchKernelGGL(pool_kernel,     dim3(4096),  dim3(256), 0, stream, x, pooled);
    hipLaunchKernelGGL(attn_up_kernel,  dim3(4096),  dim3(256), 0, stream, mt, attn_up);
    hipLaunchKernelGGL(spa_wmma_kernel, dim3(256),   dim3(256), 0, stream, ms, pooled, spa_t);
    hipLaunchKernelGGL(final_kernel,    dim3(32768), dim3(256), 0, stream, x, attn_up, spa_t, out);
}